// SizeFilter_28793460752639
// MI455X (gfx1250) — compile-verified
//
#include <hip/hip_runtime.h>
#include <stdint.h>

// Batched 3x3 Kalman filter update (F=I, H=I), B ~ 2e6.
// Memory-bound (AI ~ 0.9 flop/byte) -> stream at HBM rate via CDNA5 async
// global<->LDS b128 copies; compute 3x3 adjugate inverse in registers.

#define TILE 256

// Low 32 bits of a generic pointer to LDS == LDS byte offset (aperture in hi bits).
__device__ __forceinline__ uint32_t lds_addr(const void* p) {
    return (uint32_t)(uintptr_t)p;
}

__device__ __forceinline__ void async_load_b128(uint32_t lds, const char* g) {
    asm volatile("global_load_async_to_lds_b128 %0, %1, off"
                 :: "v"(lds), "v"(g) : "memory");
}
__device__ __forceinline__ void async_load_b32(uint32_t lds, const char* g) {
    asm volatile("global_load_async_to_lds_b32 %0, %1, off"
                 :: "v"(lds), "v"(g) : "memory");
}
__device__ __forceinline__ void async_store_b128(char* g, uint32_t lds) {
    asm volatile("global_store_async_from_lds_b128 %0, %1, off"
                 :: "v"(g), "v"(lds) : "memory");
}
__device__ __forceinline__ void async_store_b32(char* g, uint32_t lds) {
    asm volatile("global_store_async_from_lds_b32 %0, %1, off"
                 :: "v"(g), "v"(lds) : "memory");
}
__device__ __forceinline__ void wait_asynccnt0() {
    asm volatile("s_wait_asynccnt 0x0" ::: "memory");
}

// Copy nbytes from global g -> LDS s, cooperatively, dense 16B chunks per lane.
__device__ __forceinline__ void stage_in(float* s, const float* g, int nbytes,
                                         int tid, int nthreads) {
    const char* gc = (const char*)g;
    uint32_t sb = lds_addr(s);
    int chunks = nbytes >> 4;
    for (int c = tid; c < chunks; c += nthreads)
        async_load_b128(sb + ((uint32_t)c << 4), gc + ((size_t)c << 4));
    int rem = (nbytes & 15) >> 2;           // leftover dwords (generality)
    int done = chunks << 4;
    if (tid < rem)
        async_load_b32(sb + done + (tid << 2), gc + done + (tid << 2));
}

// Copy nbytes from LDS s -> global g, cooperatively, dense 16B chunks per lane.
__device__ __forceinline__ void stage_out(const float* s, float* g, int nbytes,
                                          int tid, int nthreads) {
    char* gc = (char*)g;
    uint32_t sb = lds_addr(s);
    int chunks = nbytes >> 4;
    for (int c = tid; c < chunks; c += nthreads)
        async_store_b128(gc + ((size_t)c << 4), sb + ((uint32_t)c << 4));
    int rem = (nbytes & 15) >> 2;
    int done = chunks << 4;
    if (tid < rem)
        async_store_b32(gc + done + (tid << 2), sb + done + (tid << 2));
}

__global__ __launch_bounds__(TILE) void kf3_kernel(
    const float* __restrict__ x, const float* __restrict__ P,
    const float* __restrict__ z, const float* __restrict__ R,
    const float* __restrict__ Q, float* __restrict__ out, int n) {
    // LDS layout (floats): P[9*T] R[9*T] Q[9*T] x[3*T] z[3*T] | xo[3*T] Po[9*T]
    __shared__ __attribute__((aligned(16))) float sm[TILE * 45];
    float* sP  = sm;
    float* sR  = sm + 9  * TILE;
    float* sQ  = sm + 18 * TILE;
    float* sx  = sm + 27 * TILE;
    float* sz  = sm + 30 * TILE;
    float* sxo = sm + 33 * TILE;
    float* sPo = sm + 36 * TILE;

    const int tid = threadIdx.x;
    const long long base = (long long)blockIdx.x * TILE;
    int nelem = (int)((long long)n - base);
    if (nelem > TILE) nelem = TILE;

    // ---- async stage inputs into LDS (dense, coalesced b128 per lane) ----
    stage_in(sP, P + base * 9, nelem * 36, tid, TILE);
    stage_in(sR, R + base * 9, nelem * 36, tid, TILE);
    stage_in(sQ, Q + base * 9, nelem * 36, tid, TILE);
    stage_in(sx, x + base * 3, nelem * 12, tid, TILE);
    stage_in(sz, z + base * 3, nelem * 12, tid, TILE);
    wait_asynccnt0();      // each wave drains its own async loads
    __syncthreads();       // then all waves' data is visible

    // ---- per-element 3x3 Kalman update ----
    if (tid < nelem) {
        float pp[9], S[9];
#pragma unroll
        for (int i = 0; i < 9; ++i) {
            pp[i] = sP[tid * 9 + i] + sQ[tid * 9 + i];   // P_pred = P + Q
            S[i]  = pp[i] + sR[tid * 9 + i];             // S = P_pred + R
        }
        // adjugate inverse of S (row-major S[r*3+c])
        float c00 = S[4] * S[8] - S[5] * S[7];
        float c01 = S[5] * S[6] - S[3] * S[8];
        float c02 = S[3] * S[7] - S[4] * S[6];
        float det = S[0] * c00 + S[1] * c01 + S[2] * c02;
        float id  = 1.0f / det;
        float si[9];
        si[0] = c00 * id;
        si[3] = c01 * id;
        si[6] = c02 * id;
        si[1] = (S[2] * S[7] - S[1] * S[8]) * id;
        si[4] = (S[0] * S[8] - S[2] * S[6]) * id;
        si[7] = (S[1] * S[6] - S[0] * S[7]) * id;
        si[2] = (S[1] * S[5] - S[2] * S[4]) * id;
        si[5] = (S[2] * S[3] - S[0] * S[5]) * id;
        si[8] = (S[0] * S[4] - S[1] * S[3]) * id;

        // K = P_pred * inv(S)
        float K[9];
#pragma unroll
        for (int r = 0; r < 3; ++r)
#pragma unroll
            for (int c = 0; c < 3; ++c)
                K[r * 3 + c] = pp[r * 3 + 0] * si[c]
                             + pp[r * 3 + 1] * si[3 + c]
                             + pp[r * 3 + 2] * si[6 + c];

        float xv0 = sx[tid * 3 + 0], xv1 = sx[tid * 3 + 1], xv2 = sx[tid * 3 + 2];
        float in0 = sz[tid * 3 + 0] - xv0;
        float in1 = sz[tid * 3 + 1] - xv1;
        float in2 = sz[tid * 3 + 2] - xv2;
        sxo[tid * 3 + 0] = xv0 + K[0] * in0 + K[1] * in1 + K[2] * in2;
        sxo[tid * 3 + 1] = xv1 + K[3] * in0 + K[4] * in1 + K[5] * in2;
        sxo[tid * 3 + 2] = xv2 + K[6] * in0 + K[7] * in1 + K[8] * in2;

        // P_upd = (I - K) * P_pred
        float A[9];
#pragma unroll
        for (int i = 0; i < 9; ++i)
            A[i] = ((i == 0) | (i == 4) | (i == 8) ? 1.0f : 0.0f) - K[i];
#pragma unroll
        for (int r = 0; r < 3; ++r)
#pragma unroll
            for (int c = 0; c < 3; ++c)
                sPo[tid * 9 + r * 3 + c] = A[r * 3 + 0] * pp[c]
                                         + A[r * 3 + 1] * pp[3 + c]
                                         + A[r * 3 + 2] * pp[6 + c];
    }
    __syncthreads();       // results in LDS visible to all waves

    // ---- async stream results LDS -> global (dense b128 per lane) ----
    stage_out(sxo, out + base * 3, nelem * 12, tid, TILE);
    stage_out(sPo, out + (long long)3 * n + base * 9, nelem * 36, tid, TILE);
    // S_ENDPGM performs an implicit wait-idle, draining ASYNCcnt.
}

extern "C" void kernel_launch(void* const* d_in, const int* in_sizes, int n_in,
                              void* d_out, int out_size, void* d_ws, size_t ws_size,
                              hipStream_t stream) {
    const float* x = (const float*)d_in[0];
    const float* P = (const float*)d_in[1];
    const float* z = (const float*)d_in[2];
    const float* R = (const float*)d_in[3];
    const float* Q = (const float*)d_in[4];
    float* out = (float*)d_out;
    const int n = in_sizes[0] / 3;            // B
    const int grid = (n + TILE - 1) / TILE;
    kf3_kernel<<<grid, TILE, 0, stream>>>(x, P, z, R, Q, out, n);
}